// Weight_predictor_issam_w_local_3358664425827
// MI455X (gfx1250) — compile-verified
//
#include <hip/hip_runtime.h>
#include <hip/hip_bf16.h>

typedef __attribute__((ext_vector_type(2))) float v2f;
typedef __attribute__((ext_vector_type(8))) float v8f;

#define HF 256
#define WF 256
#define C  256
#define NPATCH 1024   // 32x32 patches
#define WMASK 2048

// ---------------------------------------------------------------------------
// 1) down_mask: bilinear 2048->256, align_corners=False, scale=8
//    src = 8*i + 3.5 -> 0.5/0.5 blend of pixels (8i+3, 8i+4).
//    Read rows as coalesced b128 and select elements 3,4 of each 8-group.
// ---------------------------------------------------------------------------
__global__ void k_downsample_mask(const float* __restrict__ mask,
                                  float* __restrict__ dm) {
  const int i = blockIdx.x;    // 0..255 output row
  const int j = threadIdx.x;   // 0..255 output col
  const float* r0 = mask + (size_t)(8 * i + 3) * WMASK + 8 * j;
  const float* r1 = r0 + WMASK;
  const float4 a0 = *(const float4*)(r0);
  const float4 a1 = *(const float4*)(r0 + 4);
  const float4 b0 = *(const float4*)(r1);
  const float4 b1 = *(const float4*)(r1 + 4);
  dm[i * WF + j] = 0.25f * (a0.w + a1.x + b0.w + b1.x);
}

// ---------------------------------------------------------------------------
// 2) Streaming pass over fea (64 MB) with b128 loads.
//    Grid: 1024 blocks = (channel, quarter); each block covers 64 rows.
//    fm[c][p]   = mean over 8x8 patch of fea*dm        (p = px*32+py)
//    spart[c,q] = partial sum of fea*dm over the quarter
//    tpart[c,q] = partial sum of fea    over the quarter
//    Thread layout: rgrp = t>>6 (row in group of 4), wseg = t&63 (float4 seg)
// ---------------------------------------------------------------------------
__global__ void k_reduce_fea(const float* __restrict__ fea,
                             const float* __restrict__ dm,
                             float* __restrict__ fm,
                             float* __restrict__ spart,
                             float* __restrict__ tpart) {
  const int c = blockIdx.x >> 2;       // channel 0..255
  const int q = blockIdx.x & 3;        // quarter: bands q*8 .. q*8+7
  const int t = threadIdx.x;           // 0..255
  const int rgrp = t >> 6;             // 0..3
  const int wseg = t & 63;             // float4 segment along W
  const int w0 = wseg * 4;
  const float* feac = fea + (size_t)c * (HF * WF);

  __shared__ float sbuf[256];
  __shared__ float rs[256];
  __shared__ float rt[256];

  float ssum = 0.f, tsum = 0.f;
  for (int b = 0; b < 8; ++b) {
    const int band = q * 8 + b;        // patch row px
    float pacc = 0.f;
    #pragma unroll
    for (int half = 0; half < 2; ++half) {
      const int h = band * 8 + half * 4 + rgrp;   // covers all 8 rows over rgrp x half
      const float4 v = *(const float4*)(feac + h * WF + w0);
      const float4 m = *(const float4*)(dm + h * WF + w0);
      const float d0 = v.x * m.x + v.y * m.y;
      const float d1 = v.z * m.z + v.w * m.w;
      pacc += d0 + d1;
      ssum += d0 + d1;
      tsum += v.x + v.y + v.z + v.w;
    }
    sbuf[t] = pacc;
    __syncthreads();
    if (t < 32) {                      // t = patch column py; patch = segs {2py,2py+1} x 4 rgrps
      float s = 0.f;
      #pragma unroll
      for (int g = 0; g < 4; ++g)
        s += sbuf[g * 64 + t * 2] + sbuf[g * 64 + t * 2 + 1];
      fm[c * NPATCH + band * 32 + t] = s * (1.0f / 64.0f);
    }
    __syncthreads();
  }

  rs[t] = ssum; rt[t] = tsum;
  __syncthreads();
  for (int off = 128; off > 0; off >>= 1) {
    if (t < off) { rs[t] += rs[t + off]; rt[t] += rt[t + off]; }
    __syncthreads();
  }
  if (t == 0) {
    spart[c * 4 + q] = rs[0];
    tpart[c * 4 + q] = rt[0];
  }
}

// ---------------------------------------------------------------------------
// 3) fg/bg: fg = W@smean + b*mbar ; bg = W@(tmean-smean) + b*(1-mbar)
//    One block, 256 threads (thread = output channel). Also reduces mean(dm)
//    and folds the 4 per-quarter partials into channel means.
// ---------------------------------------------------------------------------
__global__ void k_fgbg(const float* __restrict__ dm,
                       const float* __restrict__ spart,
                       const float* __restrict__ tpart,
                       const float* __restrict__ conv_w,
                       const float* __restrict__ conv_b,
                       float* __restrict__ fg, float* __restrict__ bg) {
  __shared__ float red[256];
  __shared__ float sS[256];
  __shared__ float sD[256];
  const int t = threadIdx.x;

  float acc = 0.f;
  for (int i = t; i < HF * WF; i += 256) acc += dm[i];
  red[t] = acc;

  const float s = (spart[t * 4 + 0] + spart[t * 4 + 1] +
                   spart[t * 4 + 2] + spart[t * 4 + 3]) * (1.0f / 65536.0f);
  const float tt = (tpart[t * 4 + 0] + tpart[t * 4 + 1] +
                    tpart[t * 4 + 2] + tpart[t * 4 + 3]) * (1.0f / 65536.0f);
  sS[t] = s;
  sD[t] = tt - s;
  __syncthreads();
  for (int off = 128; off > 0; off >>= 1) {
    if (t < off) red[t] += red[t + off];
    __syncthreads();
  }
  const float mbar = red[0] * (1.0f / 65536.0f);

  float f = conv_b[t] * mbar;
  float g = conv_b[t] * (1.0f - mbar);
  const float* wrow = conv_w + (size_t)t * C;
  for (int k = 0; k < C; ++k) {
    const float w = wrow[k];
    f += w * sS[k];
    g += w * sD[k];
  }
  fg[t] = f; bg[t] = g;
}

// ---------------------------------------------------------------------------
// 4) Fold fc1: c1[j] = fc1_b[j] + fc1_w[j,0:256]@fg + fc1_w[j,256:512]@bg
//              Weff[j][k] = fc1_w[j,512+k] + fc1_w[j,768+k]   (fm appears twice)
// ---------------------------------------------------------------------------
__global__ void k_prep_fc1(const float* __restrict__ fc1_w,
                           const float* __restrict__ fc1_b,
                           const float* __restrict__ fg,
                           const float* __restrict__ bg,
                           float* __restrict__ Weff,
                           float* __restrict__ c1) {
  const int j = blockIdx.x * blockDim.x + threadIdx.x;   // 0..511
  if (j >= 512) return;
  const float* wrow = fc1_w + (size_t)j * 1024;
  float acc = fc1_b[j];
  for (int k = 0; k < 256; ++k)
    acc += wrow[k] * fg[k] + wrow[256 + k] * bg[k];
  c1[j] = acc;
  for (int k = 0; k < 256; ++k)
    Weff[j * 256 + k] = wrow[512 + k] + wrow[768 + k];
}

// ---------------------------------------------------------------------------
// 5) WMMA fp32 GEMM: h1[512,1024] = Weff[512,256] @ fm[256,1024] + c1
//    One wave per 16x16 D tile; K stepped by 4 with V_WMMA_F32_16X16X4_F32.
//    A (16x4) lane layout : lane<16 -> {K0,K1}, lane>=16 -> {K2,K3}, M=lane&15
//    B (4x16)  lane layout: lane<16 -> {K0,K1}, lane>=16 -> {K2,K3}, N=lane&15
//    D (16x16) lane layout: v0..v7 = M {0..7 | 8..15}, N=lane&15
// ---------------------------------------------------------------------------
__global__ void k_fc1_gemm_wmma(const float* __restrict__ Weff,
                                const float* __restrict__ fm,
                                const float* __restrict__ c1,
                                float* __restrict__ h1) {
  const int gwave = (blockIdx.x * blockDim.x + threadIdx.x) >> 5;  // 0..2047
  const int lane  = threadIdx.x & 31;
  const int mt = gwave >> 6;          // 0..31  M tile
  const int nt = gwave & 63;          // 0..63  N tile
  const int m0 = mt * 16, n0 = nt * 16;
  const int r    = lane & 15;         // M index (A) / N index (B,D)
  const int koff = (lane >> 4) * 2;   // 0 or 2

  v8f acc = {};
  const float* arow = Weff + (size_t)(m0 + r) * 256;
  for (int ks = 0; ks < 256; ks += 4) {
    v2f a, b;
    a.x = arow[ks + koff];
    a.y = arow[ks + koff + 1];
    b.x = fm[(ks + koff) * NPATCH + n0 + r];
    b.y = fm[(ks + koff + 1) * NPATCH + n0 + r];
    acc = __builtin_amdgcn_wmma_f32_16x16x4_f32(
        /*neg_a=*/false, a, /*neg_b=*/false, b,
        /*c_mod=*/(short)0, acc, /*reuse_a=*/false, /*reuse_b=*/false);
  }

  const int mbase = m0 + ((lane >> 4) << 3);
  const int n = n0 + r;
  #pragma unroll
  for (int v = 0; v < 8; ++v)
    h1[(size_t)(mbase + v) * NPATCH + n] = acc[v] + c1[mbase + v];
}

// ---------------------------------------------------------------------------
// 6) fc2 (3x512) per patch -> ori_weight_map [3,32,32] (tail of d_out)
// ---------------------------------------------------------------------------
__global__ void k_fc2(const float* __restrict__ h1,
                      const float* __restrict__ fc2_w,
                      const float* __restrict__ fc2_b,
                      float* __restrict__ ori) {
  const int p = blockIdx.x * blockDim.x + threadIdx.x;  // 0..1023
  if (p >= NPATCH) return;
  float a0 = fc2_b[0], a1 = fc2_b[1], a2 = fc2_b[2];
  for (int j = 0; j < 512; ++j) {
    const float h = h1[(size_t)j * NPATCH + p];
    a0 += fc2_w[j] * h;
    a1 += fc2_w[512 + j] * h;
    a2 += fc2_w[1024 + j] * h;
  }
  ori[p] = a0; ori[NPATCH + p] = a1; ori[2 * NPATCH + p] = a2;
}

// ---------------------------------------------------------------------------
// 7) Nearest expand 32x32 -> 2048x2048 (repeat 64x64), float4 stores
// ---------------------------------------------------------------------------
__global__ void k_expand(const float* __restrict__ ori, float* __restrict__ out) {
  const int tid = blockIdx.x * blockDim.x + threadIdx.x;  // 0..3145727
  const int idx = tid * 4;                                // [0, 3*2048*2048)
  const int o   = idx >> 22;                              // / (2048*2048)
  const int rem = idx & 4194303;
  const int h = rem >> 11;
  const int w = rem & 2047;
  const float v = ori[o * NPATCH + (h >> 6) * 32 + (w >> 6)];
  float4 val = make_float4(v, v, v, v);
  *(float4*)(out + idx) = val;
}

// ---------------------------------------------------------------------------
extern "C" void kernel_launch(void* const* d_in, const int* in_sizes, int n_in,
                              void* d_out, int out_size, void* d_ws, size_t ws_size,
                              hipStream_t stream) {
  const float* fea    = (const float*)d_in[0];
  const float* mask   = (const float*)d_in[1];
  const float* conv_w = (const float*)d_in[2];
  const float* conv_b = (const float*)d_in[3];
  const float* fc1_w  = (const float*)d_in[4];
  const float* fc1_b  = (const float*)d_in[5];
  const float* fc2_w  = (const float*)d_in[6];
  const float* fc2_b  = (const float*)d_in[7];

  float* ws = (float*)d_ws;
  float* dm    = ws;                       // 65536
  float* fm    = dm    + 65536;            // 256*1024
  float* spart = fm    + 256 * 1024;       // 1024
  float* tpart = spart + 1024;             // 1024
  float* fg    = tpart + 1024;             // 256
  float* bg    = fg    + 256;              // 256
  float* c1    = bg    + 256;              // 512
  float* Weff  = c1    + 512;              // 512*256
  float* h1    = Weff  + 512 * 256;        // 512*1024

  float* out_big = (float*)d_out;                    // [3,2048,2048]
  float* ori     = out_big + 3 * 2048 * 2048;        // [3,32,32]

  k_downsample_mask<<<256,  256, 0, stream>>>(mask, dm);
  k_reduce_fea     <<<1024, 256, 0, stream>>>(fea, dm, fm, spart, tpart);
  k_fgbg           <<<1,    256, 0, stream>>>(dm, spart, tpart, conv_w, conv_b, fg, bg);
  k_prep_fc1       <<<2,    256, 0, stream>>>(fc1_w, fc1_b, fg, bg, Weff, c1);
  k_fc1_gemm_wmma  <<<256,  256, 0, stream>>>(Weff, fm, c1, h1);   // 2048 waves
  k_fc2            <<<4,    256, 0, stream>>>(h1, fc2_w, fc2_b, ori);
  k_expand         <<<12288, 256, 0, stream>>>(ori, out_big);
}